// HierarchyGNN_43052752175234
// MI455X (gfx1250) — compile-verified
//
#include <hip/hip_runtime.h>

// Problem constants (match reference)
#define NN      12288
#define EE      393216
#define D_IN    256
#define HH      256
#define D_OUT   128
#define N_CLS   8
#define N_LAYERS 3

typedef __attribute__((ext_vector_type(2))) float v2f;
typedef __attribute__((ext_vector_type(8))) float v8f;
typedef __attribute__((ext_vector_type(4))) int   v4i;

#if (__has_builtin(__builtin_amdgcn_global_load_async_to_lds_b128) && \
     __has_builtin(__builtin_amdgcn_s_wait_asynccnt))
#define USE_ASYNC_COPY 1
#else
#define USE_ASYNC_COPY 0
#endif

__device__ __forceinline__ void async_copy_b128(const float* g, float* l) {
#if USE_ASYNC_COPY
    // builtin params are v4i* in device/shared address spaces; generic pointers
    // with the correct pointee type convert (same rule HIP headers rely on).
    __builtin_amdgcn_global_load_async_to_lds_b128((v4i*)(void*)g, (v4i*)(void*)l, 0, 0);
#else
    (void)g; (void)l;
#endif
}

// ---------------------------------------------------------------------------
// Generic Linear: Y[N, OUT] = act(X[N, IN] @ W[OUT, IN]^T + bias)
// IN multiple of 64, OUT multiple of 16, N multiple of 64.
// block = 128 threads (4 waves); each wave computes a 16x16 output tile with
// V_WMMA_F32_16X16X4_F32 (exact fp32). K staged through double-buffered LDS
// via GLOBAL_LOAD_ASYNC_TO_LDS_B128 (ASYNCcnt-tracked, no VGPR round-trip).
// ---------------------------------------------------------------------------
template<bool RELU>
__global__ __launch_bounds__(128)
void linear_wmma(const float* __restrict__ X, const float* __restrict__ W,
                 const float* __restrict__ bias, float* __restrict__ Y,
                 int IN, int OUT) {
    __shared__ float Xs[2][64 * 68];   // 64 rows x 64 k, padded stride 68
    __shared__ float Ws[2][16 * 68];   // 16 out-cols x 64 k

    const int rowBase = blockIdx.x * 64;
    const int o0      = blockIdx.y * 16;
    const int t    = threadIdx.x;
    const int lane = t & 31;
    const int lm   = lane & 15;   // M (A) / N (B,D) index
    const int lh   = lane >> 4;   // lane-half selector
    const int r0   = (t >> 5) * 16;
    // staging coords: thread t handles quad q4 of rows rr, rr+8, ...
    const int q4 = (t & 15) * 4;
    const int rr = t >> 4;

    v8f acc = {};
    const int nChunks = IN >> 6;

    auto compute = [&](int cur) {
        const float* xs = &Xs[cur][(r0 + lm) * 68];
        const float* ws = &Ws[cur][lm * 68];
        #pragma unroll
        for (int kk = 0; kk < 64; kk += 4) {
            const int ko = kk + lh * 2;
            v2f a, b;
            a[0] = xs[ko];
            a[1] = xs[ko + 1];
            b[0] = ws[ko];
            b[1] = ws[ko + 1];
            acc = __builtin_amdgcn_wmma_f32_16x16x4_f32(false, a, false, b,
                                                        (short)0, acc, false, false);
        }
    };

#if USE_ASYNC_COPY
    // prologue: chunk 0 -> buffer 0 (10 async b128 copies per thread-chunk)
    #pragma unroll
    for (int i = 0; i < 8; ++i)
        async_copy_b128(X + (size_t)(rowBase + i * 8 + rr) * IN + q4,
                        &Xs[0][(i * 8 + rr) * 68 + q4]);
    #pragma unroll
    for (int i = 0; i < 2; ++i)
        async_copy_b128(W + (size_t)(o0 + i * 8 + rr) * IN + q4,
                        &Ws[0][(i * 8 + rr) * 68 + q4]);

    for (int ch = 0; ch < nChunks; ++ch) {
        const int cur = ch & 1;
        if (ch + 1 < nChunks) {
            const int kb = (ch + 1) << 6;
            #pragma unroll
            for (int i = 0; i < 8; ++i)
                async_copy_b128(X + (size_t)(rowBase + i * 8 + rr) * IN + kb + q4,
                                &Xs[cur ^ 1][(i * 8 + rr) * 68 + q4]);
            #pragma unroll
            for (int i = 0; i < 2; ++i)
                async_copy_b128(W + (size_t)(o0 + i * 8 + rr) * IN + kb + q4,
                                &Ws[cur ^ 1][(i * 8 + rr) * 68 + q4]);
            // 10 next-chunk copies in flight; in-order completion => current done
            __builtin_amdgcn_s_wait_asynccnt(10);
        } else {
            __builtin_amdgcn_s_wait_asynccnt(0);
        }
        __syncthreads();   // all waves' copies for chunk ch complete
        compute(cur);
        __syncthreads();   // all waves done reading before buffer reuse
    }
#else
    // fallback: two-phase register staging (loads pipeline under one wait)
    for (int ch = 0; ch < nChunks; ++ch) {
        const int kb = ch << 6;
        float4 vx[8], vw[2];
        #pragma unroll
        for (int i = 0; i < 8; ++i)
            vx[i] = *(const float4*)(X + (size_t)(rowBase + i * 8 + rr) * IN + kb + q4);
        #pragma unroll
        for (int i = 0; i < 2; ++i)
            vw[i] = *(const float4*)(W + (size_t)(o0 + i * 8 + rr) * IN + kb + q4);
        __syncthreads();
        #pragma unroll
        for (int i = 0; i < 8; ++i) *(float4*)&Xs[0][(i * 8 + rr) * 68 + q4] = vx[i];
        #pragma unroll
        for (int i = 0; i < 2; ++i) *(float4*)&Ws[0][(i * 8 + rr) * 68 + q4] = vw[i];
        __syncthreads();
        compute(0);
    }
#endif

    // D layout: lane -> N = lane%16, vgpr r -> M = r + (lane/16)*8
    const float bv = bias[o0 + lm];
    #pragma unroll
    for (int r = 0; r < 8; ++r) {
        float v = acc[r] + bv;
        if (RELU) v = v > 0.f ? v : 0.f;
        Y[(size_t)(rowBase + r0 + lh * 8 + r) * OUT + o0 + lm] = v;
    }
}

// ---------------------------------------------------------------------------
// Graph build: dedupe edges via N*N bitmap (reproduces .set(1.0) semantics),
// degree = row sum of (A + A^T), CSR fill.
// ---------------------------------------------------------------------------
__global__ void zero_u32(unsigned* __restrict__ p, long long n) {
    long long i = (long long)blockIdx.x * blockDim.x + threadIdx.x;
    long long stride = (long long)gridDim.x * blockDim.x;
    for (; i < n; i += stride) p[i] = 0u;
}

__global__ void dedupe_edges(const int* __restrict__ ei, unsigned* __restrict__ bitmap,
                             int* __restrict__ usrc, int* __restrict__ udst,
                             int* __restrict__ cnt, int* __restrict__ deg) {
    int e = blockIdx.x * blockDim.x + threadIdx.x;
    if (e >= EE) return;
    int s = ei[e];
    int d = ei[EE + e];
    unsigned long long bit = (unsigned long long)s * NN + (unsigned long long)d;
    unsigned mask = 1u << (unsigned)(bit & 31ull);
    unsigned old = atomicOr(&bitmap[bit >> 5], mask);
    if (!(old & mask)) {
        int k = atomicAdd(cnt, 1);
        usrc[k] = s;
        udst[k] = d;
        atomicAdd(&deg[s], 1);   // out-degree of s
        atomicAdd(&deg[d], 1);   // in-degree of d (A^T side)
    }
}

__global__ void scan_deg(const int* __restrict__ deg, int* __restrict__ rowptr,
                         float* __restrict__ invdeg) {
    __shared__ int part[256];
    const int PER = NN / 256;   // 48
    int t = threadIdx.x;
    int base = t * PER;
    int s = 0;
    for (int i = 0; i < PER; ++i) s += deg[base + i];
    part[t] = s;
    __syncthreads();
    if (t == 0) {
        int acc = 0;
        for (int j = 0; j < 256; ++j) { int v = part[j]; part[j] = acc; acc += v; }
        rowptr[NN] = acc;
    }
    __syncthreads();
    int acc = part[t];
    for (int i = 0; i < PER; ++i) {
        rowptr[base + i] = acc;
        acc += deg[base + i];
        invdeg[base + i] = 1.0f / ((float)deg[base + i] + 1e-8f);
    }
}

__global__ void fill_csr(const int* __restrict__ usrc, const int* __restrict__ udst,
                         const int* __restrict__ cnt, const int* __restrict__ rowptr,
                         int* __restrict__ fillc, int* __restrict__ col) {
    int e = blockIdx.x * blockDim.x + threadIdx.x;
    if (e >= *cnt) return;
    int s = usrc[e], d = udst[e];
    col[rowptr[s] + atomicAdd(&fillc[s], 1)] = d;
    col[rowptr[d] + atomicAdd(&fillc[d], 1)] = s;
}

// y[i,:] = x[i,:] + invdeg[i] * sum_{j in row(i)} x[j,:]   (fused residual+SpMM)
// one wave per node; x rows are L2-resident (12.6 MB << 192 MB L2);
// 2-wide unroll keeps two independent gather streams in flight.
__global__ __launch_bounds__(128)
void aggregate(const float* __restrict__ x, const int* __restrict__ rowptr,
               const int* __restrict__ col, const float* __restrict__ invdeg,
               float* __restrict__ y) {
    int wv = threadIdx.x >> 5;
    int lane = threadIdx.x & 31;
    int node = blockIdx.x * 4 + wv;
    float acc0[8] = {0.f, 0.f, 0.f, 0.f, 0.f, 0.f, 0.f, 0.f};
    float acc1[8] = {0.f, 0.f, 0.f, 0.f, 0.f, 0.f, 0.f, 0.f};
    int beg = rowptr[node], end = rowptr[node + 1];
    int p = beg;
    for (; p + 2 <= end; p += 2) {
        const float* x0 = x + (size_t)col[p] * HH;
        const float* x1 = x + (size_t)col[p + 1] * HH;
        #pragma unroll
        for (int j = 0; j < 8; ++j) {
            acc0[j] += x0[j * 32 + lane];
            acc1[j] += x1[j * 32 + lane];
        }
    }
    if (p < end) {
        const float* x0 = x + (size_t)col[p] * HH;
        #pragma unroll
        for (int j = 0; j < 8; ++j) acc0[j] += x0[j * 32 + lane];
    }
    float s = invdeg[node];
    const float* xi = x + (size_t)node * HH;
    float* yo = y + (size_t)node * HH;
    #pragma unroll
    for (int j = 0; j < 8; ++j)
        yo[j * 32 + lane] = xi[j * 32 + lane] + s * (acc0[j] + acc1[j]);
}

// hierarchy classifier layer 2: [N,64] @ [8,64]^T + b  (OUT=8, tiny -> VALU)
__global__ void hc2_kernel(const float* __restrict__ x, const float* __restrict__ w,
                           const float* __restrict__ b, float* __restrict__ out) {
    int tid = blockIdx.x * blockDim.x + threadIdx.x;
    if (tid >= NN * N_CLS) return;
    int n = tid >> 3, c = tid & 7;
    const float* xr = x + (size_t)n * 64;
    const float* wr = w + (size_t)c * 64;
    float s = b[c];
    #pragma unroll 8
    for (int k = 0; k < 64; ++k) s += xr[k] * wr[k];
    out[tid] = s;
}

// global_features = mean over nodes of node_embeddings (gf pre-zeroed)
__global__ void mean_kernel(const float* __restrict__ emb, float* __restrict__ gf) {
    int c = threadIdx.x;  // 128 threads = D_OUT columns
    float s = 0.f;
    for (int n = blockIdx.x; n < NN; n += gridDim.x)
        s += emb[(size_t)n * D_OUT + c];
    atomicAdd(&gf[c], s * (1.0f / (float)NN));
}

// ---------------------------------------------------------------------------
extern "C" void kernel_launch(void* const* d_in, const int* in_sizes, int n_in,
                              void* d_out, int out_size, void* d_ws, size_t ws_size,
                              hipStream_t stream) {
    (void)in_sizes; (void)n_in; (void)out_size; (void)ws_size;

    const float* nodef  = (const float*)d_in[0];
    const int*   edges  = (const int*)d_in[1];
    const float* enc_w1 = (const float*)d_in[2];
    const float* enc_b1 = (const float*)d_in[3];
    const float* enc_w2 = (const float*)d_in[4];
    const float* enc_b2 = (const float*)d_in[5];
    const float* gin_w  = (const float*)d_in[6];
    const float* gin_b  = (const float*)d_in[7];
    const float* gl_w   = (const float*)d_in[8];   // [3,256,256]
    const float* gl_b   = (const float*)d_in[9];   // [3,256]
    const float* gout_w = (const float*)d_in[10];
    const float* gout_b = (const float*)d_in[11];
    const float* proj_w = (const float*)d_in[12];
    const float* proj_b = (const float*)d_in[13];
    const float* hc_w1  = (const float*)d_in[14];
    const float* hc_b1  = (const float*)d_in[15];
    const float* hc_w2  = (const float*)d_in[16];
    const float* hc_b2  = (const float*)d_in[17];

    float* out_emb    = (float*)d_out;                   // [N, 128]
    float* out_logits = out_emb + (size_t)NN * D_OUT;    // [N, 8]
    float* out_gf     = out_logits + (size_t)NN * N_CLS; // [128]

    // workspace layout (4-byte words; region sizes keep 16B alignment)
    const long long NH      = (long long)NN * HH;        // 3,145,728
    const long long NNWORDS = ((long long)NN * NN) >> 5; // 4,718,592
    float*    buf0   = (float*)d_ws;
    float*    buf1   = buf0 + NH;
    float*    buf2   = buf1 + NH;
    unsigned* bitmap = (unsigned*)(buf2 + NH);
    int*      usrc   = (int*)(bitmap + NNWORDS);
    int*      udst   = usrc + EE;
    int*      col    = udst + EE;                        // up to 2E entries
    int*      deg    = col + 2 * EE;
    int*      fillc  = deg + NN;
    float*    invdeg = (float*)(fillc + NN);
    int*      rowptr = (int*)(invdeg + NN);              // N+1
    int*      cnt    = rowptr + NN + 1;

    // --- zero graph-build state + global-feature accumulator ---
    zero_u32<<<4096, 256, 0, stream>>>(bitmap, NNWORDS);
    zero_u32<<<(NN + 255) / 256, 256, 0, stream>>>((unsigned*)deg, NN);
    zero_u32<<<(NN + 255) / 256, 256, 0, stream>>>((unsigned*)fillc, NN);
    zero_u32<<<1, 32, 0, stream>>>((unsigned*)cnt, 1);
    zero_u32<<<1, 128, 0, stream>>>((unsigned*)out_gf, D_OUT);

    // --- graph build ---
    dedupe_edges<<<(EE + 255) / 256, 256, 0, stream>>>(edges, bitmap, usrc, udst, cnt, deg);
    scan_deg<<<1, 256, 0, stream>>>(deg, rowptr, invdeg);
    fill_csr<<<(EE + 255) / 256, 256, 0, stream>>>(usrc, udst, cnt, rowptr, fillc, col);

    const dim3 blk(128);
    const dim3 gH(NN / 64, HH / 16);

    // --- node encoder ---
    linear_wmma<true ><<<gH, blk, 0, stream>>>(nodef, enc_w1, enc_b1, buf0, D_IN, HH);
    linear_wmma<false><<<gH, blk, 0, stream>>>(buf0, enc_w2, enc_b2, buf1, HH, HH);

    // --- SimpleGNN ---
    linear_wmma<true ><<<gH, blk, 0, stream>>>(buf1, gin_w, gin_b, buf0, HH, HH);
    for (int i = 0; i < N_LAYERS; ++i) {
        aggregate<<<NN / 4, 128, 0, stream>>>(buf0, rowptr, col, invdeg, buf2);
        linear_wmma<true><<<gH, blk, 0, stream>>>(buf2, gl_w + (size_t)i * HH * HH,
                                                  gl_b + (size_t)i * HH, buf0, HH, HH);
    }
    linear_wmma<false><<<gH, blk, 0, stream>>>(buf0, gout_w, gout_b, buf1, HH, HH);

    // --- output projection + classifier head ---
    linear_wmma<true><<<dim3(NN / 64, D_OUT / 16), blk, 0, stream>>>(
        buf1, proj_w, proj_b, out_emb, HH, D_OUT);
    linear_wmma<true><<<dim3(NN / 64, 64 / 16), blk, 0, stream>>>(
        out_emb, hc_w1, hc_b1, buf2, D_OUT, 64);
    hc2_kernel<<<(NN * N_CLS + 255) / 256, 256, 0, stream>>>(buf2, hc_w2, hc_b2, out_logits);

    // --- global mean pooling ---
    mean_kernel<<<128, D_OUT, 0, stream>>>(out_emb, out_gf);
}